// ADDSLoss_43447889167182
// MI455X (gfx1250) — compile-verified
//
#include <hip/hip_runtime.h>
#include <math.h>

// ---------------------------------------------------------------------------
// ADD-S loss on MI455X (gfx1250): pairwise-min-distance via V_WMMA_F32_16X16X4_F32
// d2[m,n] = ||p_m||^2 + (A_m . B_n), A=[px,py,pz,1], B=[-2gx,-2gy,-2gz,||g||^2]
// ---------------------------------------------------------------------------

typedef __attribute__((ext_vector_type(2))) float v2f;
typedef __attribute__((ext_vector_type(8))) float v8f;

#define BATCH 16
#define MPTS  4096          // model points
#define MTILES (MPTS / 16)  // 256 m-tiles per batch
#define UNITS (BATCH * MTILES) // 4096 wave work-units

// ---------------------------------------------------------------------------
// Kernel 1: transform model points, pack WMMA-ready A rows / B columns.
//   predA[b*M+m] = {px, py, pz, 1}
//   gtB  [b*M+n] = {-2gx, -2gy, -2gz, gx^2+gy^2+gz^2}
//   pn   [b*M+m] = px^2+py^2+pz^2
// ---------------------------------------------------------------------------
__global__ void adds_pack_kernel(const float* __restrict__ pred_R,
                                 const float* __restrict__ pred_t,
                                 const float* __restrict__ gt_R,
                                 const float* __restrict__ gt_t,
                                 const float* __restrict__ model,
                                 float4* __restrict__ predA,
                                 float4* __restrict__ gtB,
                                 float*  __restrict__ pn) {
    int idx = blockIdx.x * blockDim.x + threadIdx.x;   // 0 .. B*M-1
    int b = idx >> 12;            // M = 4096
    int m = idx & (MPTS - 1);

    float x = model[m * 3 + 0];
    float y = model[m * 3 + 1];
    float z = model[m * 3 + 2];

    // pred[b,m,j] = sum_k model[m,k] * R[b,j,k] + t[b,j]
    const float* R = pred_R + b * 9;
    const float* T = pred_t + b * 3;
    float px = fmaf(R[0], x, fmaf(R[1], y, fmaf(R[2], z, T[0])));
    float py = fmaf(R[3], x, fmaf(R[4], y, fmaf(R[5], z, T[1])));
    float pz = fmaf(R[6], x, fmaf(R[7], y, fmaf(R[8], z, T[2])));

    R = gt_R + b * 9;
    T = gt_t + b * 3;
    float gx = fmaf(R[0], x, fmaf(R[1], y, fmaf(R[2], z, T[0])));
    float gy = fmaf(R[3], x, fmaf(R[4], y, fmaf(R[5], z, T[1])));
    float gz = fmaf(R[6], x, fmaf(R[7], y, fmaf(R[8], z, T[2])));

    predA[idx] = make_float4(px, py, pz, 1.0f);
    pn[idx]    = fmaf(px, px, fmaf(py, py, pz * pz));
    gtB[idx]   = make_float4(-2.0f * gx, -2.0f * gy, -2.0f * gz,
                             fmaf(gx, gx, fmaf(gy, gy, gz * gz)));
}

// ---------------------------------------------------------------------------
// Kernel 2: one wave per (batch, m-tile of 16 pred rows). Sweep all 4096 gt
// points in 16-column tiles with V_WMMA_F32_16X16X4_F32, running elementwise
// min, then cross-lane min + sqrt + wave sum -> partial[unit].
// ---------------------------------------------------------------------------
__global__ void __launch_bounds__(256)
adds_min_kernel(const float4* __restrict__ predA,
                const float4* __restrict__ gtB,
                const float*  __restrict__ pn,
                float* __restrict__ partial) {
    const int wave = threadIdx.x >> 5;
    const int lane = threadIdx.x & 31;
    const int unit = blockIdx.x * (blockDim.x >> 5) + wave;  // 0..4095
    const int b     = unit >> 8;           // 256 m-tiles per batch
    const int mtile = unit & (MTILES - 1);
    const int m0    = mtile << 4;

    const int  r  = lane & 15;
    const bool hi = lane >= 16;

    // A fragment: 16x4 f32. Lanes 0-15: row m0+r, K=0,1; lanes 16-31: K=2,3.
    const float4 pa = predA[(b << 12) + m0 + r];
    v2f afrag;
    afrag.x = hi ? pa.z : pa.x;
    afrag.y = hi ? pa.w : pa.y;

    const float4* gB = gtB + (b << 12);

    v8f mn0 = 1e30f;   // splat
    v8f mn1 = 1e30f;
    const v8f cz = {}; // zero C

    // 4096 gt points, two 16-col tiles per iteration (independent min chains)
    for (int n0 = 0; n0 < MPTS; n0 += 32) {
        const float4 g0 = gB[n0 + r];
        const float4 g1 = gB[n0 + 16 + r];
        v2f b0, b1;
        b0.x = hi ? g0.z : g0.x;  b0.y = hi ? g0.w : g0.y;
        b1.x = hi ? g1.z : g1.x;  b1.y = hi ? g1.w : g1.y;

        // D[m,n] = -2 p.g + ||g||^2
        v8f d0 = __builtin_amdgcn_wmma_f32_16x16x4_f32(
            false, afrag, false, b0, (short)0, cz, false, false);
        v8f d1 = __builtin_amdgcn_wmma_f32_16x16x4_f32(
            false, afrag, false, b1, (short)0, cz, false, false);

        mn0 = __builtin_elementwise_min(mn0, d0);
        mn1 = __builtin_elementwise_min(mn1, d1);
    }
    v8f mn = __builtin_elementwise_min(mn0, mn1);

    // Min across the 16 lanes of each half (columns of the C/D layout).
#pragma unroll
    for (int off = 8; off >= 1; off >>= 1) {
        v8f o;
#pragma unroll
        for (int v = 0; v < 8; ++v) o[v] = __shfl_xor(mn[v], off, 32);
        mn = __builtin_elementwise_min(mn, o);
    }

    // Rows 0-7 live in lanes 0-15 (components 0-7); rows 8-15 in lanes 16-31.
    float dsum = 0.0f;
    if (r < 8) {
        float q = mn[0];
#pragma unroll
        for (int v = 1; v < 8; ++v) if (r == v) q = mn[v];
        const int row = m0 + r + (hi ? 8 : 0);
        const float d2 = pn[(b << 12) + row] + q;
        dsum = sqrtf(fmaxf(d2, 0.0f));
    }
#pragma unroll
    for (int off = 16; off >= 1; off >>= 1)
        dsum += __shfl_xor(dsum, off, 32);

    if (lane == 0) partial[unit] = dsum;
}

// ---------------------------------------------------------------------------
// Kernel 3: deterministic fixed-order tree reduction of 4096 partials -> mean.
// ---------------------------------------------------------------------------
__global__ void adds_reduce_kernel(const float* __restrict__ partial,
                                   float* __restrict__ out) {
    __shared__ float s[256];
    float v = 0.0f;
    for (int i = threadIdx.x; i < UNITS; i += 256) v += partial[i];
    s[threadIdx.x] = v;
    __syncthreads();
#pragma unroll
    for (int stride = 128; stride > 0; stride >>= 1) {
        if (threadIdx.x < stride) s[threadIdx.x] += s[threadIdx.x + stride];
        __syncthreads();
    }
    if (threadIdx.x == 0) out[0] = s[0] * (1.0f / (float)(BATCH * MPTS));
}

// ---------------------------------------------------------------------------
extern "C" void kernel_launch(void* const* d_in, const int* in_sizes, int n_in,
                              void* d_out, int out_size, void* d_ws, size_t ws_size,
                              hipStream_t stream) {
    const float* pred_R = (const float*)d_in[0];  // [16,3,3]
    const float* pred_t = (const float*)d_in[1];  // [16,3]
    const float* gt_R   = (const float*)d_in[2];  // [16,3,3]
    const float* gt_t   = (const float*)d_in[3];  // [16,3]
    const float* model  = (const float*)d_in[4];  // [4096,3]
    float* out = (float*)d_out;

    // Workspace layout (all 16B-aligned offsets):
    //   predA : B*M float4  (1 MiB)
    //   gtB   : B*M float4  (1 MiB)
    //   pn    : B*M float   (256 KiB)
    //   part  : UNITS float (16 KiB)
    char* ws = (char*)d_ws;
    float4* predA = (float4*)(ws);
    float4* gtB   = (float4*)(ws + (size_t)BATCH * MPTS * sizeof(float4));
    float*  pn    = (float*) (ws + (size_t)2 * BATCH * MPTS * sizeof(float4));
    float*  part  = (float*) (ws + (size_t)2 * BATCH * MPTS * sizeof(float4)
                                 + (size_t)BATCH * MPTS * sizeof(float));

    // 1) pack: 65536 points, 256 threads/block
    adds_pack_kernel<<<(BATCH * MPTS) / 256, 256, 0, stream>>>(
        pred_R, pred_t, gt_R, gt_t, model, predA, gtB, pn);

    // 2) WMMA min sweep: 4096 wave-units, 8 waves (256 threads) per block
    adds_min_kernel<<<UNITS / 8, 256, 0, stream>>>(predA, gtB, pn, part);

    // 3) deterministic mean
    adds_reduce_kernel<<<1, 256, 0, stream>>>(part, out);
}